// SFTRL_CCFM_23227183136964
// MI455X (gfx1250) — compile-verified
//
#include <hip/hip_runtime.h>
#include <math.h>

// Problem constants (from reference)
#define D        1024
#define TWO_M    64
#define M_HALF   32
#define NDATA    20000
#define ETA_F    1e-4f
#define THRES_F  1e-12f

#define NTHREADS 512
#define NSWEEP   8
#define COLPAD   1025   // column stride in floats: 1025 % 64 == 1 -> conflict-free

typedef float v2f __attribute__((ext_vector_type(2)));
typedef float v8f __attribute__((ext_vector_type(8)));

__device__ __forceinline__ v8f wmma_f32_16x16x4(v2f a, v2f b, v8f c) {
  // D = A(16x4,f32) * B(4x16,f32) + C(16x16,f32)
  return __builtin_amdgcn_wmma_f32_16x16x4_f32(false, a, false, b, (short)0, c,
                                               false, false);
}

// ---- LDS layout (float offsets) ----
#define OFF_B      0         // 64 * 1025         column-major B, padded
#define OFF_A      65600     // 1024              alpha row
#define OFF_PART   66624     // 512               GEMV partials
#define OFF_BPA    67136     // 64                B^T alpha
#define OFF_G      67200     // 64*65             Gram G; later reused as U_sorted
#define OFF_U      71360     // 64*65             Jacobi eigenvector accumulator
#define OFF_CS     75520     // 32                rotation cos
#define OFF_SS     75552     // 32                rotation sin
#define OFF_SSORT  75584     // 64                sorted eigenvalues
#define OFF_FACTOR 75648     // 64                per-column rebuild factor
#define OFF_PERM   75712     // 64 ints           round-robin permutation
#define OFF_P      75776     // 32 ints
#define OFF_Q      75808     // 32 ints
#define OFF_ORDER  75840     // 64 ints           eigen sort order
#define OFF_CTRL   75904     // 8                 [0]=rc [1]=upd [2]=wcol [3]=newrc, f[4]=coef
#define SMEM_FLOATS 75912    // 303,648 bytes  (< 320 KB WGP LDS)

__global__ __launch_bounds__(NTHREADS, 1)
void sftrl_ccfm_kernel(const float* __restrict__ X,
                       const float* __restrict__ y,
                       float* __restrict__ preds)
{
  extern __shared__ float sm[];
  float* sB    = sm + OFF_B;
  float* sA    = sm + OFF_A;
  float* sPart = sm + OFF_PART;
  float* sBPa  = sm + OFF_BPA;
  float* sG    = sm + OFF_G;
  float* sU    = sm + OFF_U;
  float* sC    = sm + OFF_CS;
  float* sS    = sm + OFF_SS;
  float* sEv   = sm + OFF_SSORT;
  float* sFac  = sm + OFF_FACTOR;
  int*   sPerm = (int*)(sm + OFF_PERM);
  int*   sP    = (int*)(sm + OFF_P);
  int*   sQ    = (int*)(sm + OFF_Q);
  int*   sOrd  = (int*)(sm + OFF_ORDER);
  int*   sCi   = (int*)(sm + OFF_CTRL);
  float* sCf   = sm + OFF_CTRL + 4;

  const int t    = threadIdx.x;
  const int lane = t & 31;
  const int wave = t >> 5;

  // B = 0, rc = 0
  for (int i = t; i < TWO_M * COLPAD; i += NTHREADS) sB[i] = 0.0f;
  if (t == 0) sCi[0] = 0;
  __syncthreads();

  for (int step = 0; step < NDATA; ++step) {
    // ---- stage alpha row into LDS; prefetch next row (global_prefetch_b8) ----
    const float* row = X + (size_t)step * D;
    sA[t]            = row[t];
    sA[t + NTHREADS] = row[t + NTHREADS];
    if (step + 1 < NDATA) {
      const float* nrow = X + (size_t)(step + 1) * D;
      __builtin_prefetch(nrow + t, 0, 1);
      __builtin_prefetch(nrow + t + NTHREADS, 0, 1);
    }
    __syncthreads();

    // ---- BPa = B^T * alpha  (VALU GEMV; N=1 so WMMA would waste 15/16) ----
    {
      const int j = t & 63, seg = t >> 6;          // 64 cols x 8 segments
      const float* col = sB + j * COLPAD + seg * 128;
      const float* av  = sA + seg * 128;
      float acc = 0.0f;
      #pragma unroll 8
      for (int c = 0; c < 128; ++c) acc = fmaf(col[c], av[c], acc);
      sPart[j * 8 + seg] = acc;
    }
    __syncthreads();
    if (t < TWO_M) {
      float s = 0.0f;
      #pragma unroll
      for (int k = 0; k < 8; ++k) s += sPart[t * 8 + k];
      sBPa[t] = s;
    }
    __syncthreads();

    // ---- scalar = ||BPa||^2 ; decide update (wave 0) ----
    if (t < 32) {
      float v = sBPa[t] * sBPa[t] + sBPa[t + 32] * sBPa[t + 32];
      #pragma unroll
      for (int off = 16; off > 0; off >>= 1) v += __shfl_xor(v, off, 32);
      if (t == 0) {
        preds[step] = v;
        float sign = 2.0f * (v - y[step]);
        if (sign <= 0.0f) {
          sCi[1] = 1;
          sCi[2] = sCi[0] + 1;                     // reference increments BEFORE write
          sCf[0] = sqrtf(-ETA_F * sign);
        } else {
          sCi[1] = 0;
        }
      }
    }
    __syncthreads();

    if (sCi[1]) {                                  // uniform branch (LDS flag)
      const int   col  = sCi[2];
      const float coef = sCf[0];
      sB[col * COLPAD + t]            = coef * sA[t];
      sB[col * COLPAD + t + NTHREADS] = coef * sA[t + NTHREADS];
      __syncthreads();

      if (col == TWO_M - 1) {
        // ================= shrink =================
        // (1) G = B^T B  -- 16 waves, one 16x16 tile each, v_wmma_f32_16x16x4_f32
        {
          const int ta = wave >> 2, tb = wave & 3;
          const int l = lane & 15, koff = (lane < 16) ? 0 : 2;
          const float* pa = sB + (ta * 16 + l) * COLPAD + koff;  // A = B^T rows
          const float* pb = sB + (tb * 16 + l) * COLPAD + koff;  // B = B cols
          v8f acc = {};
          for (int k0 = 0; k0 < D; k0 += 4) {
            v2f a, b;
            a.x = pa[k0]; a.y = pa[k0 + 1];
            b.x = pb[k0]; b.y = pb[k0 + 1];
            acc = wmma_f32_16x16x4(a, b, acc);
          }
          const int rbase = ta * 16 + ((lane < 16) ? 0 : 8);
          #pragma unroll
          for (int v = 0; v < 8; ++v)
            sG[(rbase + v) * 65 + tb * 16 + l] = acc[v];
        }
        // U = I ; round-robin permutation init
        for (int e = t; e < TWO_M * TWO_M; e += NTHREADS) {
          int i = e >> 6, j = e & 63;
          sU[i * 65 + j] = (i == j) ? 1.0f : 0.0f;
        }
        if (t < TWO_M) sPerm[t] = t;
        __syncthreads();

        // (2) parallel two-sided cyclic Jacobi on G (32 disjoint rotations/round)
        for (int round = 0; round < 63 * NSWEEP; ++round) {
          if (t < 32) {
            int p = sPerm[t], q = sPerm[63 - t];
            if (p > q) { int tmp = p; p = q; q = tmp; }
            float app = sG[p * 65 + p], aqq = sG[q * 65 + q], apq = sG[p * 65 + q];
            float c = 1.0f, s = 0.0f;
            if (fabsf(apq) > 1e-30f) {
              float tau = (aqq - app) / (2.0f * apq);
              float tt  = 1.0f / (tau + sqrtf(1.0f + tau * tau));
              c = 1.0f / sqrtf(1.0f + tt * tt);
              s = tt * c;
            }
            sP[t] = p; sQ[t] = q; sC[t] = c; sS[t] = s;
          }
          __syncthreads();
          if (t == 0) {                            // rotate schedule (snapshot taken)
            int last = sPerm[63];
            for (int i = 63; i >= 2; --i) sPerm[i] = sPerm[i - 1];
            sPerm[1] = last;
          }
          {  // row update: A = J^T G
            const int k = t >> 4, j0 = t & 15;
            const int p = sP[k], q = sQ[k];
            const float c = sC[k], s = sS[k];
            #pragma unroll
            for (int jj = 0; jj < 4; ++jj) {
              int j = j0 + jj * 16;
              float gp = sG[p * 65 + j], gq = sG[q * 65 + j];
              sG[p * 65 + j] = c * gp - s * gq;
              sG[q * 65 + j] = s * gp + c * gq;
            }
          }
          __syncthreads();
          {  // column update: G = A J, and U = U J
            const int k = t >> 4, i0 = t & 15;
            const int p = sP[k], q = sQ[k];
            const float c = sC[k], s = sS[k];
            #pragma unroll
            for (int ii = 0; ii < 4; ++ii) {
              int i = i0 + ii * 16;
              float gp = sG[i * 65 + p], gq = sG[i * 65 + q];
              sG[i * 65 + p] = c * gp - s * gq;
              sG[i * 65 + q] = s * gp + c * gq;
              float up = sU[i * 65 + p], uq = sU[i * 65 + q];
              sU[i * 65 + p] = c * up - s * uq;
              sU[i * 65 + q] = s * up + c * uq;
            }
          }
          __syncthreads();
        }

        // (3) sort eigenvalues desc, threshold, per-column factors (serial, rare)
        if (t == 0) {
          for (int j = 0; j < TWO_M; ++j) { sEv[j] = sG[j * 65 + j]; sOrd[j] = j; }
          for (int a = 0; a < TWO_M; ++a) {
            int best = a;
            for (int b = a + 1; b < TWO_M; ++b)
              if (sEv[b] > sEv[best]) best = b;
            float tv = sEv[a]; sEv[a] = sEv[best]; sEv[best] = tv;
            int ti = sOrd[a]; sOrd[a] = sOrd[best]; sOrd[best] = ti;
          }
          int nnz = 0;
          for (int j = 0; j < TWO_M; ++j) {
            if (sEv[j] <= THRES_F) sEv[j] = 0.0f;
            if (sEv[j] > 0.0f) nnz++;
          }
          const int ge = (nnz >= M_HALF);
          const float Sm = sEv[M_HALF];
          for (int j = 0; j < TWO_M; ++j) {
            float scale = ge ? ((j < M_HALF - 1) ? sqrtf(fmaxf(sEv[j] - Sm, 0.0f)) : 0.0f)
                             : ((j < nnz) ? sqrtf(sEv[j]) : 0.0f);
            float invs  = (sEv[j] > 0.0f) ? (1.0f / sqrtf(sEv[j])) : 0.0f;
            sFac[j] = invs * scale;
          }
          sCi[3] = ge ? (M_HALF - 1) : nnz;
        }
        __syncthreads();
        // (4) U_sorted (reuse sG): sG[k][j] = sU[k][order[j]]
        for (int e = t; e < TWO_M * TWO_M; e += NTHREADS) {
          int i = e >> 6, j = e & 63;
          sG[i * 65 + j] = sU[i * 65 + sOrd[j]];
        }
        __syncthreads();
        // (5) B <- (B @ U_sorted) * diag(factor) via WMMA; each wave owns 4
        //     exclusive 16-row bands -> read then overwrite in place, no temp.
        {
          const int l = lane & 15, koff = (lane < 16) ? 0 : 2;
          for (int sbi = 0; sbi < 4; ++sbi) {
            const int r0 = (wave * 4 + sbi) * 16;
            v8f acc0 = {}, acc1 = {}, acc2 = {}, acc3 = {};
            for (int k0 = 0; k0 < TWO_M; k0 += 4) {
              v2f a;
              a.x = sB[(k0 + koff)     * COLPAD + r0 + l];
              a.y = sB[(k0 + koff + 1) * COLPAD + r0 + l];
              v2f b0, b1, b2, b3;
              b0.x = sG[(k0 + koff) * 65 + l];      b0.y = sG[(k0 + koff + 1) * 65 + l];
              b1.x = sG[(k0 + koff) * 65 + 16 + l]; b1.y = sG[(k0 + koff + 1) * 65 + 16 + l];
              b2.x = sG[(k0 + koff) * 65 + 32 + l]; b2.y = sG[(k0 + koff + 1) * 65 + 32 + l];
              b3.x = sG[(k0 + koff) * 65 + 48 + l]; b3.y = sG[(k0 + koff + 1) * 65 + 48 + l];
              acc0 = wmma_f32_16x16x4(a, b0, acc0);
              acc1 = wmma_f32_16x16x4(a, b1, acc1);
              acc2 = wmma_f32_16x16x4(a, b2, acc2);
              acc3 = wmma_f32_16x16x4(a, b3, acc3);
            }
            const int mbase = (lane < 16) ? 0 : 8;
            #pragma unroll
            for (int v = 0; v < 8; ++v) {
              const int rr = r0 + mbase + v;
              sB[(l)      * COLPAD + rr] = acc0[v] * sFac[l];
              sB[(16 + l) * COLPAD + rr] = acc1[v] * sFac[16 + l];
              sB[(32 + l) * COLPAD + rr] = acc2[v] * sFac[32 + l];
              sB[(48 + l) * COLPAD + rr] = acc3[v] * sFac[48 + l];
            }
          }
        }
        __syncthreads();
        if (t == 0) sCi[0] = sCi[3];               // rc = m-1 or nnz
      } else {
        if (t == 0) sCi[0] = col;                  // rc = rc + 1
      }
    }
    __syncthreads();
  }
}

extern "C" void kernel_launch(void* const* d_in, const int* in_sizes, int n_in,
                              void* d_out, int out_size, void* d_ws, size_t ws_size,
                              hipStream_t stream) {
  (void)in_sizes; (void)n_in; (void)d_ws; (void)ws_size; (void)out_size;
  const float* X = (const float*)d_in[0];   // (20000, 1024) f32
  const float* y = (const float*)d_in[1];   // (20000,)      f32
  float* preds   = (float*)d_out;           // (20000,)      f32
  sftrl_ccfm_kernel<<<dim3(1), dim3(NTHREADS), SMEM_FLOATS * sizeof(float), stream>>>(
      X, y, preds);
}